// CLIPVisionTower_80247168958810
// MI455X (gfx1250) — compile-verified
//
#include <hip/hip_runtime.h>
#include <stdint.h>

// Problem shape (fixed by reference): B=16, H=16, S=577, D=1024, NUM_COUNT=256
#define BATCH    16
#define HEADS    16
#define SEQ      577
#define DIM      1024
#define NSEL     257            // NUM_COUNT + 1
#define PER_HEAD 16             // NUM_COUNT / H
#define LPL      19             // ceil(SEQ / 32) values per lane (wave32)
#define THREADS  256
#define COPY_BLOCKS 4080

// ---------------------------------------------------------------------------
// Fused kernel:
//   blocks [0, BATCH)        : per-batch sequential top-k selection (wave32)
//   blocks [BATCH, gridDim)  : async-LDS streaming copy of hidden_states->out
// ---------------------------------------------------------------------------
__global__ void __launch_bounds__(THREADS)
fused_select_copy(const float* __restrict__ attn,
                  const float* __restrict__ hs,
                  float* __restrict__ out_hs,
                  int* __restrict__ sel,
                  size_t nchunks)
{
    // 16 CLS rows * 577 floats = 36928 B; also reused as async-copy staging.
    __shared__ float attn_lds[HEADS * SEQ];
    const int tid = threadIdx.x;

    if (blockIdx.x < BATCH) {
        // ----------------- selection for batch b -----------------
        const int b = blockIdx.x;
        const size_t head_stride = (size_t)SEQ * SEQ;
        const float* base = attn + (size_t)b * HEADS * head_stride;

        // Cooperative load of attention[b, h, 0, :] for all 16 heads.
        for (int k = tid; k < HEADS * SEQ; k += THREADS) {
            const int h = k / SEQ;
            const int i = k - h * SEQ;
            attn_lds[k] = base[(size_t)h * head_stride + i];
        }
        __syncthreads();

        if (tid < 32) {
            const int lane = tid;
            // member bit j <-> global index (lane + 32*j). Index 0 preselected.
            uint32_t memberbits = (lane == 0) ? 1u : 0u;

            for (int h = 0; h < HEADS; ++h) {
                float vals[LPL];
                const float* row = &attn_lds[h * SEQ];
                #pragma unroll
                for (int j = 0; j < LPL; ++j) {
                    const int idx = lane + 32 * j;
                    float v = -1.0f;   // attention is uniform[0,1): sentinel loses
                    if (idx < SEQ && idx != 0 && !((memberbits >> j) & 1u))
                        v = row[idx];
                    vals[j] = v;
                }
                // 16 iterative argmax picks == top-16 new indices in rank order
                for (int t = 0; t < PER_HEAD; ++t) {
                    float bv = vals[0];
                    int   bj = 0;
                    #pragma unroll
                    for (int j = 1; j < LPL; ++j)
                        if (vals[j] > bv) { bv = vals[j]; bj = j; }  // ties: smaller idx
                    int bidx = lane + 32 * bj;
                    // wave32 butterfly reduction; tie -> smaller index (argsort-stable)
                    #pragma unroll
                    for (int o = 16; o >= 1; o >>= 1) {
                        const float ov = __shfl_xor(bv, o, 32);
                        const int   oi = __shfl_xor(bidx, o, 32);
                        if (ov > bv || (ov == bv && oi < bidx)) { bv = ov; bidx = oi; }
                    }
                    // owning lane marks winner as member, masks it out
                    if ((bidx & 31) == lane) {
                        const int j = bidx >> 5;
                        vals[j] = -1.0f;
                        memberbits |= (1u << j);
                    }
                }
            }

            // Compact member set into ascending index list via ballot+popc.
            int cumbase = 0;
            #pragma unroll
            for (int j = 0; j < LPL; ++j) {
                const uint32_t bit  = (memberbits >> j) & 1u;
                const uint32_t mask = (uint32_t)__ballot((int)bit);
                if (bit) {
                    const int pos = cumbase + __popc(mask & ((1u << lane) - 1u));
                    sel[b * NSEL + pos] = lane + 32 * j;
                }
                cumbase += __popc(mask);
            }
        }
    } else {
        // ----------------- async-LDS streaming copy -----------------
        // Each lane stages 4x16B chunks through its private LDS slots using the
        // CDNA5 async global<->LDS engine (ASYNCcnt-tracked).
        uint8_t* stage = (uint8_t*)attn_lds;                  // 256*64 B = 16 KB used
        const uint32_t lds0 = (uint32_t)(uintptr_t)(stage + (size_t)tid * 64);
        const size_t gid    = (size_t)(blockIdx.x - BATCH) * THREADS + tid;
        const size_t stride = (size_t)(gridDim.x - BATCH) * THREADS;
        const uint8_t* src  = (const uint8_t*)hs;
        uint8_t* dst        = (uint8_t*)out_hs;

        for (size_t c0 = gid; c0 < nchunks; c0 += 4 * stride) {
            #pragma unroll
            for (int u = 0; u < 4; ++u) {
                const size_t c = c0 + (size_t)u * stride;
                if (c < nchunks) {
                    const uint64_t ga  = (uint64_t)(uintptr_t)(src + c * 16);
                    const uint32_t lds = lds0 + (uint32_t)(u * 16);
                    asm volatile("global_load_async_to_lds_b128 %0, %1, off"
                                 :: "v"(lds), "v"(ga) : "memory");
                }
            }
            asm volatile("s_wait_asynccnt 0" ::: "memory");
            #pragma unroll
            for (int u = 0; u < 4; ++u) {
                const size_t c = c0 + (size_t)u * stride;
                if (c < nchunks) {
                    const uint64_t gb  = (uint64_t)(uintptr_t)(dst + c * 16);
                    const uint32_t lds = lds0 + (uint32_t)(u * 16);
                    asm volatile("global_store_async_from_lds_b128 %0, %1, off"
                                 :: "v"(gb), "v"(lds) : "memory");
                }
            }
            asm volatile("s_wait_asynccnt 0" ::: "memory");
        }
    }
}

// ---------------------------------------------------------------------------
// Gather: one 4 KB row (1024 floats) per 256-thread block, float4 per thread.
// ---------------------------------------------------------------------------
__global__ void __launch_bounds__(THREADS)
gather_kernel(const float* __restrict__ hs,
              const int* __restrict__ sel,
              float* __restrict__ outq)
{
    const int row  = blockIdx.x;          // b * NSEL + j
    const int b    = row / NSEL;
    const int srow = sel[row];
    const float4* src = (const float4*)(hs + ((size_t)b * SEQ + srow) * DIM);
    float4*       dst = (float4*)(outq + (size_t)row * DIM);
    dst[threadIdx.x] = src[threadIdx.x];
}

extern "C" void kernel_launch(void* const* d_in, const int* in_sizes, int n_in,
                              void* d_out, int out_size, void* d_ws, size_t ws_size,
                              hipStream_t stream)
{
    (void)in_sizes; (void)n_in; (void)out_size; (void)ws_size;
    const float* hs   = (const float*)d_in[0];  // [16, 577, 1024] f32
    const float* attn = (const float*)d_in[1];  // [16, 16, 577, 577] f32
    float* out = (float*)d_out;                 // [hs | selected_query] flat
    int*   sel = (int*)d_ws;                    // [16, 257] selected indices

    const size_t nfloat  = (size_t)BATCH * SEQ * DIM;          // 9,453,568
    const size_t nchunks = nfloat * sizeof(float) / 16;        // 16B chunks

    fused_select_copy<<<BATCH + COPY_BLOCKS, THREADS, 0, stream>>>(
        attn, hs, out, sel, nchunks);
    gather_kernel<<<BATCH * NSEL, THREADS, 0, stream>>>(
        hs, sel, out + nfloat);
}